// TopologicalVAELoss_81003083202862
// MI455X (gfx1250) — compile-verified
//
#include <hip/hip_runtime.h>
#include <math.h>

#define B_SZ   1024
#define D_DATA 16384
#define D_LAT  256
#define N_GRID 256
#define EPSF   1e-12f

typedef __attribute__((ext_vector_type(2))) float v2f;
typedef __attribute__((ext_vector_type(4))) float v4f;
typedef __attribute__((ext_vector_type(8))) float v8f;

// ---- workspace layout (float offsets) ----
#define SQ_OFF     0                          // 1024 sq norms
#define RECON_OFF  1024                       // 2048 recon block partials
#define RECON_BLKS 2048
#define KL_OFF     (RECON_OFF + RECON_BLKS)   // 64 kl block partials
#define KL_BLKS    64
#define DEATHS_OFF (KL_OFF + KL_BLKS)         // 1023 MST edge weights
#define DIST_OFF   (DEATHS_OFF + 1024)        // 1024*1024 distance matrix

// ---------------- squared norms of representation rows ----------------
__global__ void __launch_bounds__(256) sqnorm_kernel(const float* __restrict__ z,
                                                     float* __restrict__ sq) {
    int row = blockIdx.x * blockDim.x + threadIdx.x;
    if (row >= B_SZ) return;
    const v4f* zr = (const v4f*)(z + (size_t)row * D_LAT);
    float acc = 0.f;
#pragma unroll 4
    for (int k = 0; k < D_LAT / 4; ++k) {
        v4f v = zr[k];
        acc += v.x * v.x + v.y * v.y + v.z * v.z + v.w * v.w;
    }
    sq[row] = acc;
}

// ---------------- Gram matrix via V_WMMA_F32_16X16X4_F32 -> distances ----------------
// One wave (32 lanes) computes one 16x16 tile of D = sqrt(clip(sq_i + sq_j - 2*G, 0) + eps).
// A-operand (16x4 f32): lane l (half h) holds {z[m0+l][k0+2h], z[m0+l][k0+2h+1]} -> float2 load.
// B-operand (4x16 f32): identical shape with the n-panel (B = z^T columns).
__global__ void __launch_bounds__(32) dist_wmma_kernel(const float* __restrict__ z,
                                                       const float* __restrict__ sq,
                                                       float* __restrict__ dist) {
    const int lane = threadIdx.x;
    const int half = lane >> 4;
    const int l    = lane & 15;
    const int m0   = blockIdx.y * 16;
    const int n0   = blockIdx.x * 16;

    const float* arow = z + (size_t)(m0 + l) * D_LAT + 2 * half;
    const float* brow = z + (size_t)(n0 + l) * D_LAT + 2 * half;

    v8f c = {};
#pragma unroll 8
    for (int k0 = 0; k0 < D_LAT; k0 += 4) {
        v2f a = *(const v2f*)(arow + k0);
        v2f b = *(const v2f*)(brow + k0);
        // D = A x B + C, f32 in / f32 accum (16x16x4)
        c = __builtin_amdgcn_wmma_f32_16x16x4_f32(false, a, false, b,
                                                  (short)0, c, false, false);
    }

    const int   n   = n0 + l;
    const float sqn = sq[n];
#pragma unroll
    for (int r = 0; r < 8; ++r) {
        int   m  = m0 + r + 8 * half;      // C/D layout: VGPR r, halves hold M=r / M=r+8
        float d2 = sq[m] + sqn - 2.0f * c[r];
        dist[(size_t)m * B_SZ + n] = sqrtf(fmaxf(d2, 0.0f) + EPSF);
    }
}

// ---------------- recon MSE partial sums ----------------
// 128 MB of single-use streaming: the dominant term (~5.5 us at 23.3 TB/s).
// Non-temporal loads keep this stream from displacing the z/dist working set.
__global__ void __launch_bounds__(256) recon_kernel(const float* __restrict__ r,
                                                    const float* __restrict__ x,
                                                    float* __restrict__ partial) {
    __shared__ float s[256];
    const int tid = threadIdx.x;
    const int n4  = (B_SZ * (D_DATA / 4));               // 4,194,304 float4s
    const int stride = gridDim.x * blockDim.x;
    const v4f* r4 = (const v4f*)r;
    const v4f* x4 = (const v4f*)x;
    float acc = 0.f;
    for (int i = blockIdx.x * blockDim.x + tid; i < n4; i += stride) {
        v4f a = __builtin_nontemporal_load(r4 + i);
        v4f b = __builtin_nontemporal_load(x4 + i);
        float dx = a.x - b.x, dy = a.y - b.y, dz = a.z - b.z, dw = a.w - b.w;
        acc += dx * dx + dy * dy + dz * dz + dw * dw;
    }
    s[tid] = acc;
    __syncthreads();
    for (int off = 128; off > 0; off >>= 1) {
        if (tid < off) s[tid] += s[tid + off];
        __syncthreads();
    }
    if (tid == 0) partial[blockIdx.x] = s[0];
}

// ---------------- KL partial sums ----------------
__global__ void __launch_bounds__(256) kl_kernel(const float* __restrict__ mu,
                                                 const float* __restrict__ logvar,
                                                 float* __restrict__ partial) {
    __shared__ float s[256];
    const int tid = threadIdx.x;
    const int n = B_SZ * D_LAT;
    const int stride = gridDim.x * blockDim.x;
    float acc = 0.f;
    for (int i = blockIdx.x * blockDim.x + tid; i < n; i += stride) {
        float m  = __builtin_nontemporal_load(mu + i);
        float lv = __builtin_nontemporal_load(logvar + i);
        acc += 1.0f + lv - m * m - expf(lv);
    }
    s[tid] = acc;
    __syncthreads();
    for (int off = 128; off > 0; off >>= 1) {
        if (tid < off) s[tid] += s[tid + off];
        __syncthreads();
    }
    if (tid == 0) partial[blockIdx.x] = s[0];
}

// ---------------- Prim's MST: 1023 serial argmin+relax steps ----------------
__global__ void __launch_bounds__(1024) prim_kernel(const float* __restrict__ dist,
                                                    float* __restrict__ deaths) {
    __shared__ float s_val[1024];
    __shared__ int   s_idx[1024];
    __shared__ float s_mind[1024];
    __shared__ int   s_tree[1024];
    const int tid = threadIdx.x;

    s_mind[tid] = dist[tid];            // row 0
    s_tree[tid] = (tid == 0) ? 1 : 0;
    __syncthreads();

    for (int step = 0; step < B_SZ - 1; ++step) {
        s_val[tid] = s_tree[tid] ? __builtin_inff() : s_mind[tid];
        s_idx[tid] = tid;
        __syncthreads();
        // argmin with first-index tie-break (matches jnp.argmin)
        for (int off = 512; off > 0; off >>= 1) {
            if (tid < off) {
                float v1 = s_val[tid],      v2 = s_val[tid + off];
                int   i1 = s_idx[tid],      i2 = s_idx[tid + off];
                if (v2 < v1 || (v2 == v1 && i2 < i1)) { s_val[tid] = v2; s_idx[tid] = i2; }
            }
            __syncthreads();
        }
        int j = s_idx[0];
        if (tid == 0) deaths[step] = s_val[0];
        if (tid == j) s_tree[tid] = 1;
        float dij = dist[(size_t)j * B_SZ + tid];
        s_mind[tid] = fminf(s_mind[tid], dij);
        __syncthreads();
    }
}

// ---------------- silhouette + final combine ----------------
__global__ void __launch_bounds__(256) final_kernel(const float* __restrict__ ws,
                                                    float* __restrict__ out) {
    __shared__ float s_a[256];
    __shared__ float s_b[256];
    const int t = threadIdx.x;
    const float* deaths = ws + DEATHS_OFF;

    // max and sum of deaths
    float mx = -__builtin_inff(), sm = 0.f;
    for (int j = t; j < B_SZ - 1; j += 256) {
        float d = deaths[j];
        mx = fmaxf(mx, d);
        sm += d;
    }
    s_a[t] = mx; s_b[t] = sm;
    __syncthreads();
    for (int off = 128; off > 0; off >>= 1) {
        if (t < off) { s_a[t] = fmaxf(s_a[t], s_a[t + off]); s_b[t] += s_b[t + off]; }
        __syncthreads();
    }
    const float tmax = s_a[0];
    const float wsum = s_b[0];
    __syncthreads();

    // phi(t_i), one grid point per thread; t_i = (i/255)*tmax
    float tt  = ((float)t / (float)(N_GRID - 1)) * tmax;
    float acc = 0.f;
    for (int j = 0; j < B_SZ - 1; ++j) {
        float d = deaths[j];
        acc += d * fmaxf(0.0f, fminf(tt, d - tt));
    }
    s_a[t] = acc / (wsum + EPSF);
    __syncthreads();
    for (int off = 128; off > 0; off >>= 1) {
        if (t < off) s_a[t] += s_a[t + off];
        __syncthreads();
    }
    const float topo = s_a[0] / (float)N_GRID;
    __syncthreads();

    // fold recon / kl partials
    float racc = 0.f;
    for (int j = t; j < RECON_BLKS; j += 256) racc += ws[RECON_OFF + j];
    s_a[t] = racc;
    s_b[t] = (t < KL_BLKS) ? ws[KL_OFF + t] : 0.f;
    __syncthreads();
    for (int off = 128; off > 0; off >>= 1) {
        if (t < off) { s_a[t] += s_a[t + off]; s_b[t] += s_b[t + off]; }
        __syncthreads();
    }
    if (t == 0) {
        float recon = s_a[0] / ((float)B_SZ * (float)D_DATA);
        float kl    = -0.5f * s_b[0] / ((float)B_SZ * (float)D_LAT);
        float total = recon + 0.1f * kl + 0.5f * topo;
        out[0] = total; out[1] = recon; out[2] = kl; out[3] = topo;
    }
}

extern "C" void kernel_launch(void* const* d_in, const int* in_sizes, int n_in,
                              void* d_out, int out_size, void* d_ws, size_t ws_size,
                              hipStream_t stream) {
    const float* recon_x = (const float*)d_in[0];
    const float* x       = (const float*)d_in[1];
    const float* mu      = (const float*)d_in[2];
    const float* logvar  = (const float*)d_in[3];
    const float* z       = (const float*)d_in[4];
    float* out = (float*)d_out;
    float* ws  = (float*)d_ws;

    sqnorm_kernel<<<(B_SZ + 255) / 256, 256, 0, stream>>>(z, ws + SQ_OFF);

    dim3 gtiles(B_SZ / 16, B_SZ / 16);   // 64 x 64 tiles, one wave each
    dist_wmma_kernel<<<gtiles, 32, 0, stream>>>(z, ws + SQ_OFF, ws + DIST_OFF);

    recon_kernel<<<RECON_BLKS, 256, 0, stream>>>(recon_x, x, ws + RECON_OFF);
    kl_kernel<<<KL_BLKS, 256, 0, stream>>>(mu, logvar, ws + KL_OFF);

    prim_kernel<<<1, 1024, 0, stream>>>(ws + DIST_OFF, ws + DEATHS_OFF);
    final_kernel<<<1, 256, 0, stream>>>(ws, out);
}